// DeltaNet_31877247271454
// MI455X (gfx1250) — compile-verified
//
#include <hip/hip_runtime.h>
#include <math.h>

// ---------------------------------------------------------------------------
// DeltaNet chunkwise delta-rule for MI455X (gfx1250, wave32).
//  b=4, h=4, L=8192, d=128, CHUNK=32.
//  Pass 1: per-chunk prep (normalize, A, triangular inverse, u, w, attn)
//          -> fully parallel over 4096 chunks, WMMA f32 16x16x4.
//  Pass 2: sequential scan over 256 chunks, parallel over 16 (b,h) x 4
//          dv-strips of 32 columns (S columns are independent) = 64 blocks.
//          Per-chunk u0/attn staged via Tensor Data Mover (TENSORcnt),
//          next-chunk q/k/w warmed with global_prefetch.
// Workspace: u (64MB) + w (64MB) + attn (16MB) = 144MB fp32 (L2-resident).
// ---------------------------------------------------------------------------

typedef float v2f __attribute__((ext_vector_type(2)));
typedef float v8f __attribute__((ext_vector_type(8)));

#define BH      16
#define SEQ     8192
#define D       128
#define C       32
#define NCHUNK  256          // SEQ / C
#define DVB     32           // dv strip width in pass 2
#define NB      (D / DVB)    // 4 strips

#if defined(__HIP_DEVICE_COMPILE__) && defined(__gfx1250__) && \
    __has_builtin(__builtin_amdgcn_tensor_load_to_lds) &&      \
    __has_builtin(__builtin_amdgcn_s_wait_tensorcnt)
#define HAVE_TDM 1
#else
#define HAVE_TDM 0
#endif

__device__ __forceinline__ v8f wmma4(v2f a, v2f b, v8f c) {
  // V_WMMA_F32_16X16X4_F32: D = A(16x4) x B(4x16) + C(16x16), fp32 throughout.
  return __builtin_amdgcn_wmma_f32_16x16x4_f32(
      /*neg_a=*/false, a, /*neg_b=*/false, b,
      /*c_mod=*/(short)0, c, /*reuse_a=*/false, /*reuse_b=*/false);
}

// A-fragment from row-major src[m][k] (leading dim ld), 16x4 tile @ (rb, k0).
// Lane l: m = rb + (l&15), k = k0 + (l>>4)*2 -> contiguous pair.
__device__ __forceinline__ v2f afrag(const float* s, int ld, int rb, int k0) {
  int l = threadIdx.x & 31;
  const float* p = s + (rb + (l & 15)) * ld + k0 + ((l >> 4) << 1);
  v2f r; r[0] = p[0]; r[1] = p[1]; return r;
}
// A-fragment where A[m][k] = s[k][m] (transposed source, row-major s, ld).
__device__ __forceinline__ v2f afragT(const float* s, int ld, int rb, int k0) {
  int l = threadIdx.x & 31;
  int m = rb + (l & 15);
  int k = k0 + ((l >> 4) << 1);
  v2f r; r[0] = s[k * ld + m]; r[1] = s[(k + 1) * ld + m]; return r;
}
// B-fragment from row-major src[k][n] (leading dim ld), 4x16 tile @ (k0, cb).
__device__ __forceinline__ v2f bfrag(const float* s, int ld, int k0, int cb) {
  int l = threadIdx.x & 31;
  int n = cb + (l & 15);
  int k = k0 + ((l >> 4) << 1);
  v2f r; r[0] = s[k * ld + n]; r[1] = s[(k + 1) * ld + n]; return r;
}
// B-fragment where B[k][n] = s[n][k] (transposed source) -> contiguous pair.
__device__ __forceinline__ v2f bfragT(const float* s, int ld, int k0, int cb) {
  int l = threadIdx.x & 31;
  const float* p = s + (cb + (l & 15)) * ld + k0 + ((l >> 4) << 1);
  v2f r; r[0] = p[0]; r[1] = p[1]; return r;
}
// 16x16 C/D tile: VGPR i holds row (rb + (l>>4)*8 + i), col cb + (l&15).
__device__ __forceinline__ v8f load_tile(const float* s, int ld, int rb, int cb) {
  int l = threadIdx.x & 31;
  int n = cb + (l & 15);
  int m0 = rb + ((l >> 4) << 3);
  v8f c;
#pragma unroll
  for (int i = 0; i < 8; ++i) c[i] = s[(m0 + i) * ld + n];
  return c;
}
__device__ __forceinline__ void store_tile(float* d, int ld, int rb, int cb, v8f c) {
  int l = threadIdx.x & 31;
  int n = cb + (l & 15);
  int m0 = rb + ((l >> 4) << 3);
#pragma unroll
  for (int i = 0; i < 8; ++i) d[(m0 + i) * ld + n] = c[i];
}

#if HAVE_TDM
typedef unsigned int u32x4 __attribute__((ext_vector_type(4)));
typedef int i32x4 __attribute__((ext_vector_type(4)));
typedef int i32x8 __attribute__((ext_vector_type(8)));

// Low 32 bits of a generic pointer to LDS = byte offset within the
// workgroup's LDS allocation (flat LDS aperture mapping, ISA 10.2).
__device__ __forceinline__ unsigned lds_lo32(const void* p) {
  return (unsigned)(unsigned long long)p;
}

// TDM 2-D f32 tile load: D# group0/group1 per CDNA5 ISA.
// tile_w x tile_h elements, source row stride row_stride elements.
// 6-arg builtin form (clang-23 / amdgpu-toolchain): extra i32x8 before cpol.
__device__ __forceinline__ void tdm_load_f32_2d(
    const float* gsrc, unsigned lds_byte,
    unsigned tile_w, unsigned tile_h,
    unsigned tensor_w, unsigned tensor_h,
    unsigned long long row_stride) {
  unsigned long long ga = (unsigned long long)gsrc;
  u32x4 g0;
  g0[0] = 1u;                                                 // count=1 (user)
  g0[1] = lds_byte;                                           // lds_addr
  g0[2] = (unsigned)ga;                                       // global_addr lo
  g0[3] = (unsigned)((ga >> 32) & 0x01ffffffu) | (2u << 30);  // ga[56:32]|type=2
  i32x8 g1;
  g1[0] = (int)(2u << 16);                                    // data_size=4B
  g1[1] = (int)((tensor_w & 0xffffu) << 16);                  // dim0[15:0]
  g1[2] = (int)(((tensor_w >> 16) & 0xffffu) |
                ((tensor_h & 0xffffu) << 16));                // dim0 hi|dim1 lo
  g1[3] = (int)(((tensor_h >> 16) & 0xffffu) |
                ((tile_w & 0xffffu) << 16));                  // dim1 hi|tile0
  g1[4] = (int)(tile_h & 0xffffu);                            // tile1, tile2=0
  g1[5] = (int)(unsigned)(row_stride & 0xffffffffu);          // stride0 lo
  g1[6] = (int)(unsigned)((row_stride >> 32) & 0xffffu);      // stride0 hi
  g1[7] = 0;
  i32x4 z4 = {0, 0, 0, 0};
  i32x8 z8 = {0, 0, 0, 0, 0, 0, 0, 0};
  __builtin_amdgcn_tensor_load_to_lds(g0, g1, z4, z4, z8, 0);
}
#endif

// ===========================================================================
// Pass 1: per-chunk prep.  grid = BH*NCHUNK = 4096, block = 128 (4 waves).
// ===========================================================================
__global__ __launch_bounds__(128) void deltanet_prep(
    const float* __restrict__ q, const float* __restrict__ k,
    const float* __restrict__ v, const float* __restrict__ beta,
    float* __restrict__ u_ws, float* __restrict__ w_ws,
    float* __restrict__ attn_ws) {
  __shared__ __align__(16) float qn[C * D];
  __shared__ __align__(16) float kn[C * D];
  __shared__ __align__(16) float vb[C * D];
  __shared__ __align__(16) float Am[C * C];
  __shared__ __align__(16) float betas[C];

  const int blk = blockIdx.x;                 // bh*NCHUNK + ci
  const int tid = threadIdx.x;
  const int wid = tid >> 5;
  const int l = tid & 31;
  const size_t rowbase = (size_t)blk * C;     // first seq row of this chunk

  // ---- stage & normalize: wave0->q, wave1->k, wave2->v*beta, wave3->betas
  {
    int r = l;
    if (wid < 2) {
      const float* src = (wid == 0 ? q : k) + (rowbase + r) * D;
      float* dst = (wid == 0 ? qn : kn) + r * D;
      float ss = 0.f;
#pragma unroll
      for (int i = 0; i < D / 4; ++i) {
        float4 x = ((const float4*)src)[i];
        ((float4*)dst)[i] = x;
        ss += x.x * x.x + x.y * x.y + x.z * x.z + x.w * x.w;
      }
      float rn = rsqrtf(ss + 1e-6f);
#pragma unroll
      for (int i = 0; i < D / 4; ++i) {
        float4 x = ((float4*)dst)[i];
        x.x *= rn; x.y *= rn; x.z *= rn; x.w *= rn;
        ((float4*)dst)[i] = x;
      }
    } else if (wid == 2) {
      const float* src = v + (rowbase + r) * D;
      float bv = beta[rowbase + r];
      float* dst = vb + r * D;
#pragma unroll
      for (int i = 0; i < D / 4; ++i) {
        float4 x = ((const float4*)src)[i];
        x.x *= bv; x.y *= bv; x.z *= bv; x.w *= bv;
        ((float4*)dst)[i] = x;
      }
    } else {
      betas[r] = beta[rowbase + r];
    }
  }
  __syncthreads();

  // ---- A = -strict_tril( (kn*beta) @ kn^T ): one 16x16 tile per wave ----
  {
    int tm = (wid >> 1) << 4, tn = (wid & 1) << 4;
    v8f acc = {0.f, 0.f, 0.f, 0.f, 0.f, 0.f, 0.f, 0.f};
    float bm = betas[tm + (l & 15)];          // row-scale for kb = kn*beta
    for (int k0 = 0; k0 < D; k0 += 4) {
      v2f a = afrag(kn, D, tm, k0);
      a[0] *= bm; a[1] *= bm;
      v2f b = bfragT(kn, D, k0, tn);          // B = kn^T
      acc = wmma4(a, b, acc);
    }
    int n = tn + (l & 15);
    int m0 = tm + ((l >> 4) << 3);
#pragma unroll
    for (int i = 0; i < 8; ++i) {
      int m = m0 + i;
      Am[m * C + n] = (m > n) ? -acc[i] : 0.f;
    }
  }
  __syncthreads();

  // ---- T = (forward-substitution inverse) + I, serial on wave 0 ----
  if (wid == 0) {
    for (int i = 1; i < C; ++i) {
      float acc = Am[i * C + l];
      for (int j = 0; j < i; ++j)
        acc += Am[i * C + j] * Am[j * C + l];  // rows j<i already updated
      Am[i * C + l] = acc;                     // equals original for l >= i
      asm volatile("s_wait_dscnt 0" ::: "memory");
    }
    Am[l * C + l] += 1.0f;
  }
  __syncthreads();

  // ---- u = T @ (v*beta),  w = T @ (kn*beta):  32 16x16 tiles / 4 waves ----
  for (int t = wid; t < 32; t += 4) {
    int which = t >> 4;                        // 0: u, 1: w
    int idx = t & 15;
    int tm = (idx >> 3) << 4;                  // 0 | 16
    int tn = (idx & 7) << 4;                   // 0..112
    v8f acc = {0.f, 0.f, 0.f, 0.f, 0.f, 0.f, 0.f, 0.f};
    for (int k0 = 0; k0 < C; k0 += 4) {
      v2f a = afrag(Am, C, tm, k0);
      v2f b;
      if (which == 0) {
        b = bfrag(vb, D, k0, tn);
      } else {
        b = bfrag(kn, D, k0, tn);
        int kk = k0 + ((l >> 4) << 1);
        b[0] *= betas[kk];
        b[1] *= betas[kk + 1];
      }
      acc = wmma4(a, b, acc);
    }
    float* dst = (which == 0 ? u_ws : w_ws) + (size_t)blk * (C * D);
    store_tile(dst, D, tm, tn, acc);
  }

  // ---- attn = tril(qn @ kn^T, 0) (diag inclusive): one tile per wave ----
  {
    int tm = (wid >> 1) << 4, tn = (wid & 1) << 4;
    v8f acc = {0.f, 0.f, 0.f, 0.f, 0.f, 0.f, 0.f, 0.f};
    for (int k0 = 0; k0 < D; k0 += 4)
      acc = wmma4(afrag(qn, D, tm, k0), bfragT(kn, D, k0, tn), acc);
    float* dst = attn_ws + (size_t)blk * (C * C);
    int n = tn + (l & 15);
    int m0 = tm + ((l >> 4) << 3);
#pragma unroll
    for (int i = 0; i < 8; ++i) {
      int m = m0 + i;
      dst[m * C + n] = (m >= n) ? acc[i] : 0.f;
    }
  }
}

// ===========================================================================
// Pass 2: sequential scan.  grid = BH*NB = 64, block = 128 (4 waves).
// Each block owns S[:, vb0:vb0+32] (128x32 in LDS) for one (b,h).
// ===========================================================================
__global__ __launch_bounds__(128) void deltanet_scan(
    const float* __restrict__ q, const float* __restrict__ k,
    const float* __restrict__ u_ws, const float* __restrict__ w_ws,
    const float* __restrict__ attn_ws,
    float* __restrict__ o_out, float* __restrict__ s_out) {
  __shared__ __align__(16) float Sm[D * DVB];   // state strip [dk][dv]
  __shared__ __align__(16) float qn[C * D];
  __shared__ __align__(16) float kn[C * D];
  __shared__ __align__(16) float u0[C * DVB];
  __shared__ __align__(16) float un[C * DVB];
  __shared__ __align__(16) float at[C * C];

  const int bh = blockIdx.x / NB;
  const int vb0 = (blockIdx.x % NB) * DVB;
  const int tid = threadIdx.x;
  const int wid = tid >> 5;
  const int l = tid & 31;

  for (int i = tid; i < D * DVB; i += 128) Sm[i] = 0.f;
  __syncthreads();

  for (int ci = 0; ci < NCHUNK; ++ci) {
    const size_t chunk = (size_t)bh * NCHUNK + ci;
    const size_t rowbase = chunk * C;

#if HAVE_TDM
    // ---- TDM: DMA u0 strip (32x32 of a 128-stride tensor) and attn tile
    //      into LDS while all 4 waves normalize q/k.  Wave 0 issues/waits.
    if (wid == 0) {
      tdm_load_f32_2d(u_ws + chunk * (C * D) + vb0, lds_lo32(u0),
                      DVB, C, DVB, C, (unsigned long long)D);
      tdm_load_f32_2d(attn_ws + chunk * (C * C), lds_lo32(at),
                      C, C, C, C, (unsigned long long)C);
    }
    // ---- q/k: 2 threads per row, pair-reduce the sumsq with shfl_xor ----
    {
      int rr = tid >> 1;                       // 0..63: 0-31 q rows, 32-63 k
      int half = tid & 1;
      int r = rr & 31;
      const float* src = (rr < 32 ? q : k) + (rowbase + r) * D + half * 64;
      float* dst = (rr < 32 ? qn : kn) + r * D + half * 64;
      float ss = 0.f;
#pragma unroll
      for (int i = 0; i < 16; ++i) {
        float4 x = ((const float4*)src)[i];
        ((float4*)dst)[i] = x;
        ss += x.x * x.x + x.y * x.y + x.z * x.z + x.w * x.w;
      }
      ss += __shfl_xor(ss, 1);
      float rn = rsqrtf(ss + 1e-6f);
#pragma unroll
      for (int i = 0; i < 16; ++i) {
        float4 x = ((float4*)dst)[i];
        x.x *= rn; x.y *= rn; x.z *= rn; x.w *= rn;
        ((float4*)dst)[i] = x;
      }
    }
    if (wid == 0) __builtin_amdgcn_s_wait_tensorcnt(0);
#else
    // ---- fallback staging: w0->qn, w1->kn, w2->u0 strip, w3->attn ----
    {
      int r = l;
      if (wid < 2) {
        const float* src = (wid == 0 ? q : k) + (rowbase + r) * D;
        float* dst = (wid == 0 ? qn : kn) + r * D;
        float ss = 0.f;
#pragma unroll
        for (int i = 0; i < D / 4; ++i) {
          float4 x = ((const float4*)src)[i];
          ((float4*)dst)[i] = x;
          ss += x.x * x.x + x.y * x.y + x.z * x.z + x.w * x.w;
        }
        float rn = rsqrtf(ss + 1e-6f);
#pragma unroll
        for (int i = 0; i < D / 4; ++i) {
          float4 x = ((float4*)dst)[i];
          x.x *= rn; x.y *= rn; x.z *= rn; x.w *= rn;
          ((float4*)dst)[i] = x;
        }
      } else if (wid == 2) {
        const float* src = u_ws + chunk * (C * D) + (size_t)r * D + vb0;
#pragma unroll
        for (int i = 0; i < DVB / 4; ++i)
          ((float4*)(u0 + r * DVB))[i] = ((const float4*)src)[i];
      } else {
        const float* src = attn_ws + chunk * (C * C) + (size_t)r * C;
#pragma unroll
        for (int i = 0; i < C / 4; ++i)
          ((float4*)(at + r * C))[i] = ((const float4*)src)[i];
      }
    }
#endif
    __syncthreads();

    // ---- warm next chunk's q/k/w (one 128B line per thread) ----
    if (ci + 1 < NCHUNK) {
      const size_t nxt = chunk + 1;
      __builtin_prefetch(w_ws + nxt * (C * D) + (size_t)tid * 32, 0, 1);
      __builtin_prefetch(q + nxt * (C * D) + (size_t)tid * 32, 0, 1);
      __builtin_prefetch(k + nxt * (C * D) + (size_t)tid * 32, 0, 1);
    }

    const float* wg = w_ws + chunk * (C * D);  // w streamed from global (L2)

    // ---- un = u0 - w @ S : one 16x16 tile per wave ----
    {
      int tm = (wid >> 1) << 4, tn = (wid & 1) << 4;
      v8f acc = {0.f, 0.f, 0.f, 0.f, 0.f, 0.f, 0.f, 0.f};
      for (int k0 = 0; k0 < D; k0 += 4)
        acc = wmma4(afrag(wg, D, tm, k0), bfrag(Sm, DVB, k0, tn), acc);
      int n = tn + (l & 15);
      int m0 = tm + ((l >> 4) << 3);
#pragma unroll
      for (int i = 0; i < 8; ++i) {
        int m = m0 + i;
        un[m * DVB + n] = u0[m * DVB + n] - acc[i];
      }
    }
    __syncthreads();

    // ---- o = qn @ S + attn @ un : one tile per wave, chained accumulator ----
    {
      int tm = (wid >> 1) << 4, tn = (wid & 1) << 4;
      v8f acc = {0.f, 0.f, 0.f, 0.f, 0.f, 0.f, 0.f, 0.f};
      for (int k0 = 0; k0 < D; k0 += 4)
        acc = wmma4(afrag(qn, D, tm, k0), bfrag(Sm, DVB, k0, tn), acc);
      for (int k0 = 0; k0 < C; k0 += 4)
        acc = wmma4(afrag(at, C, tm, k0), bfrag(un, DVB, k0, tn), acc);
      int n = tn + (l & 15);
      int m0 = tm + ((l >> 4) << 3);
#pragma unroll
      for (int i = 0; i < 8; ++i)
        o_out[(rowbase + m0 + i) * D + vb0 + n] = acc[i];
    }
    __syncthreads();   // all S reads done before update

    // ---- S += kn^T @ un : 16 tiles (8 dk x 2 dv) / 4 waves ----
    for (int t = wid; t < 16; t += 4) {
      int tm = (t >> 1) << 4;                  // dk tile base 0..112
      int tn = (t & 1) << 4;
      v8f c = load_tile(Sm, DVB, tm, tn);
      for (int k0 = 0; k0 < C; k0 += 4)
        c = wmma4(afragT(kn, D, tm, k0), bfrag(un, DVB, k0, tn), c);
      store_tile(Sm, DVB, tm, tn, c);
    }
    __syncthreads();
  }

  // ---- final state: s_out[bh][dk][dv] strip ----
  {
    float* dst = s_out + (size_t)bh * D * D + (size_t)tid * D + vb0;
    const float* src = Sm + tid * DVB;
#pragma unroll
    for (int i = 0; i < DVB; ++i) dst[i] = src[i];
  }
}

extern "C" void kernel_launch(void* const* d_in, const int* in_sizes, int n_in,
                              void* d_out, int out_size, void* d_ws, size_t ws_size,
                              hipStream_t stream) {
  const float* q = (const float*)d_in[0];
  const float* k = (const float*)d_in[1];
  const float* v = (const float*)d_in[2];
  const float* beta = (const float*)d_in[3];

  float* u_ws = (float*)d_ws;                                   // 64 MB
  float* w_ws = u_ws + (size_t)BH * NCHUNK * C * D;             // 64 MB
  float* attn_ws = w_ws + (size_t)BH * NCHUNK * C * D;          // 16 MB

  float* o = (float*)d_out;                                     // [b,h,L,dv]
  float* s = o + (size_t)BH * SEQ * D;                          // [b,h,dk,dv]

  deltanet_prep<<<BH * NCHUNK, 128, 0, stream>>>(q, k, v, beta, u_ws, w_ws,
                                                 attn_ws);
  deltanet_scan<<<BH * NB, 128, 0, stream>>>(q, k, u_ws, w_ws, attn_ws, o, s);
}